// GATMeta_19224273617371
// MI455X (gfx1250) — compile-verified
//
#include <hip/hip_runtime.h>

// Problem constants (match reference)
#define G_     2048
#define NPG    22
#define EPG    440
#define CH     64
#define FIN    22
#define NTOT   (G_ * NPG)        // 45056
#define E_     (G_ * EPG)        // 901120
#define ETOT_G (EPG + NPG)       // 462 edges per graph incl. self-loops
#define GN_EPS 1e-5f
#define SLOPE  0.2f

typedef float v2f __attribute__((ext_vector_type(2)));
typedef float v8f __attribute__((ext_vector_type(8)));

// ---------------------------------------------------------------------------
// Y[N,64] = A[N,K] @ W[K,64] in fp32 via V_WMMA_F32_16X16X4_F32.
// One wave computes a 32x64 output tile: 2 row-tiles x 4 col-tiles = 8 v8f
// accumulators; each B fragment is reused for both row tiles.
// K is a template constant so bound guards vanish (K=64) or apply only to the
// final k-step (K=22 -> padded to 24 in registers). Loop fully unrolled.
// Fragment layouts per CDNA5 ISA 7.12.2:
//   A 16x4 f32 : lanes 0-15 hold {K=0,K=1}, lanes 16-31 hold {K=2,K=3}, M=lane&15
//   B 4x16 f32 : lanes 0-15 hold rows {0,1}, lanes 16-31 rows {2,3}, N=lane&15
//   C/D 16x16  : VGPR v -> M = v + 8*(lane>=16), N = lane&15
// ---------------------------------------------------------------------------
template <int K>
__global__ void __launch_bounds__(32) wmma_gemm_n64(
    const float* __restrict__ A, const float* __restrict__ W,
    float* __restrict__ Y)
{
  const int lane  = threadIdx.x & 31;
  const int khalf = lane >> 4;
  const int r     = lane & 15;
  const int row0  = blockIdx.x * 32;

  v8f a00 = {}, a01 = {}, a02 = {}, a03 = {};   // rows row0 .. row0+15
  v8f a10 = {}, a11 = {}, a12 = {}, a13 = {};   // rows row0+16 .. row0+31

  constexpr int Kpad = (K + 3) & ~3;
  #pragma unroll
  for (int k = 0; k < Kpad; k += 4) {
    const int ka = k + 2 * khalf;

    v2f fa0, fa1;
    v2f b0, b1, b2, b3;
    if constexpr (Kpad == K) {
      // no tail: packed 8-byte loads, no guards
      fa0 = *(const v2f*)(A + (size_t)(row0 + r) * K + ka);
      fa1 = *(const v2f*)(A + (size_t)(row0 + 16 + r) * K + ka);
      b0.x = W[ka * CH +  0 + r];        b0.y = W[(ka + 1) * CH +  0 + r];
      b1.x = W[ka * CH + 16 + r];        b1.y = W[(ka + 1) * CH + 16 + r];
      b2.x = W[ka * CH + 32 + r];        b2.y = W[(ka + 1) * CH + 32 + r];
      b3.x = W[ka * CH + 48 + r];        b3.y = W[(ka + 1) * CH + 48 + r];
    } else if (k + 4 <= K) {
      fa0 = *(const v2f*)(A + (size_t)(row0 + r) * K + ka);
      fa1 = *(const v2f*)(A + (size_t)(row0 + 16 + r) * K + ka);
      b0.x = W[ka * CH +  0 + r];        b0.y = W[(ka + 1) * CH +  0 + r];
      b1.x = W[ka * CH + 16 + r];        b1.y = W[(ka + 1) * CH + 16 + r];
      b2.x = W[ka * CH + 32 + r];        b2.y = W[(ka + 1) * CH + 32 + r];
      b3.x = W[ka * CH + 48 + r];        b3.y = W[(ka + 1) * CH + 48 + r];
    } else {
      // tail k-step only (K=22): zero-pad K in registers
      fa0.x = (ka     < K) ? A[(size_t)(row0 + r) * K + ka    ] : 0.0f;
      fa0.y = (ka + 1 < K) ? A[(size_t)(row0 + r) * K + ka + 1] : 0.0f;
      fa1.x = (ka     < K) ? A[(size_t)(row0 + 16 + r) * K + ka    ] : 0.0f;
      fa1.y = (ka + 1 < K) ? A[(size_t)(row0 + 16 + r) * K + ka + 1] : 0.0f;
      b0.x = (ka < K) ? W[ka * CH +  0 + r] : 0.0f;
      b1.x = (ka < K) ? W[ka * CH + 16 + r] : 0.0f;
      b2.x = (ka < K) ? W[ka * CH + 32 + r] : 0.0f;
      b3.x = (ka < K) ? W[ka * CH + 48 + r] : 0.0f;
      b0.y = (ka + 1 < K) ? W[(ka + 1) * CH +  0 + r] : 0.0f;
      b1.y = (ka + 1 < K) ? W[(ka + 1) * CH + 16 + r] : 0.0f;
      b2.y = (ka + 1 < K) ? W[(ka + 1) * CH + 32 + r] : 0.0f;
      b3.y = (ka + 1 < K) ? W[(ka + 1) * CH + 48 + r] : 0.0f;
    }

    a00 = __builtin_amdgcn_wmma_f32_16x16x4_f32(false, fa0, false, b0, (short)0, a00, false, false);
    a01 = __builtin_amdgcn_wmma_f32_16x16x4_f32(false, fa0, false, b1, (short)0, a01, false, false);
    a02 = __builtin_amdgcn_wmma_f32_16x16x4_f32(false, fa0, false, b2, (short)0, a02, false, false);
    a03 = __builtin_amdgcn_wmma_f32_16x16x4_f32(false, fa0, false, b3, (short)0, a03, false, false);
    a10 = __builtin_amdgcn_wmma_f32_16x16x4_f32(false, fa1, false, b0, (short)0, a10, false, false);
    a11 = __builtin_amdgcn_wmma_f32_16x16x4_f32(false, fa1, false, b1, (short)0, a11, false, false);
    a12 = __builtin_amdgcn_wmma_f32_16x16x4_f32(false, fa1, false, b2, (short)0, a12, false, false);
    a13 = __builtin_amdgcn_wmma_f32_16x16x4_f32(false, fa1, false, b3, (short)0, a13, false, false);
  }

  const int col = lane & 15;
  #pragma unroll
  for (int v = 0; v < 8; ++v) {
    const size_t rA = row0 + v + 8 * khalf;
    const size_t rB = rA + 16;
    Y[rA * CH +  0 + col] = a00[v];
    Y[rA * CH + 16 + col] = a01[v];
    Y[rA * CH + 32 + col] = a02[v];
    Y[rA * CH + 48 + col] = a03[v];
    Y[rB * CH +  0 + col] = a10[v];
    Y[rB * CH + 16 + col] = a11[v];
    Y[rB * CH + 32 + col] = a12[v];
    Y[rB * CH + 48 + col] = a13[v];
  }
}

// ---------------------------------------------------------------------------
// One workgroup per graph: edge attention logits + leaky-relu + per-dst
// softmax + weighted aggregation + bias (+ ReLU) + GraphNorm, all in LDS.
// FINAL additionally does mean-pool + 64->2 classifier and writes d_out.
// All reductions are fixed-order scans (deterministic; no float atomics).
// ---------------------------------------------------------------------------
template <int RELU, int FINAL>
__global__ void __launch_bounds__(256) gat_graph_kernel(
    const float* __restrict__ xl, const float* __restrict__ xr,
    const int* __restrict__ e_src, const int* __restrict__ e_dst,
    const float* __restrict__ att, const float* __restrict__ bias,
    const float* __restrict__ gw, const float* __restrict__ gb,
    const float* __restrict__ gms,
    float* __restrict__ h_out,
    const float* __restrict__ linW, const float* __restrict__ linb,
    float* __restrict__ out_o, float* __restrict__ out_pooled)
{
  __shared__ __align__(16) float s_xl[NPG * CH];
  __shared__ __align__(16) float s_xr[NPG * CH];
  __shared__ __align__(16) float s_acc[NPG * CH];
  __shared__ float         s_alpha[ETOT_G];   // logits, then alpha
  __shared__ unsigned char s_sl[ETOT_G];
  __shared__ unsigned char s_dl[ETOT_G];
  __shared__ float s_att[CH];
  __shared__ float s_m[NPG], s_z[NPG];
  __shared__ float s_msmean[CH], s_rstd[CH];
  __shared__ float s_pool[CH];

  const int g        = blockIdx.x;
  const int tid      = threadIdx.x;
  const int nodeBase = g * NPG;

  // stage this graph's xl/xr tiles with b128 loads (22*64 floats = 352 float4)
  {
    const float4* gxl = (const float4*)(xl + (size_t)nodeBase * CH);
    const float4* gxr = (const float4*)(xr + (size_t)nodeBase * CH);
    float4* lxl = (float4*)s_xl;
    float4* lxr = (float4*)s_xr;
    for (int i = tid; i < NPG * CH / 4; i += 256) {
      lxl[i] = gxl[i];
      lxr[i] = gxr[i];
    }
  }
  if (tid < CH) s_att[tid] = att[tid];
  for (int e = tid; e < ETOT_G; e += 256) {
    int sl, dl;
    if (e < EPG) {
      sl = e_src[(size_t)g * EPG + e] - nodeBase;
      dl = e_dst[(size_t)g * EPG + e] - nodeBase;
    } else {
      sl = dl = e - EPG;   // PyG add_self_loops
    }
    s_sl[e] = (unsigned char)sl;
    s_dl[e] = (unsigned char)dl;
  }
  __syncthreads();

  // 1) edge logits: s = att . leaky_relu(xl[src] + xr[dst])
  for (int e = tid; e < ETOT_G; e += 256) {
    const int sl = s_sl[e], dl = s_dl[e];
    float s = 0.0f;
    #pragma unroll 8
    for (int c = 0; c < CH; ++c) {
      float h = s_xl[sl * CH + c] + s_xr[dl * CH + c];
      h = (h > 0.0f) ? h : SLOPE * h;
      s += s_att[c] * h;
    }
    s_alpha[e] = s;
  }
  __syncthreads();

  // 2) per-destination softmax stats (fixed-order scan per node)
  if (tid < NPG) {
    float m = -3.402823466e38f;
    for (int e = 0; e < ETOT_G; ++e)
      if (s_dl[e] == tid) m = fmaxf(m, s_alpha[e]);
    float z = 0.0f;
    for (int e = 0; e < ETOT_G; ++e)
      if (s_dl[e] == tid) z += __expf(s_alpha[e] - m);
    s_m[tid] = m;
    s_z[tid] = z;
  }
  __syncthreads();
  for (int e = tid; e < ETOT_G; e += 256) {
    const int dl = s_dl[e];
    s_alpha[e] = __expf(s_alpha[e] - s_m[dl]) / (s_z[dl] + 1e-16f);
  }
  __syncthreads();

  // 3) aggregate: acc[n][c] = sum_{e: dst==n} alpha_e * xl[src_e][c]; +bias; ReLU
  for (int t = tid; t < NPG * (CH / 4); t += 256) {
    const int n  = t >> 4;
    const int c4 = (t & 15) << 2;
    float4 acc = make_float4(0.f, 0.f, 0.f, 0.f);
    for (int e = 0; e < ETOT_G; ++e) {
      if (s_dl[e] == n) {
        const float  a = s_alpha[e];
        const float4 v = *reinterpret_cast<const float4*>(&s_xl[s_sl[e] * CH + c4]);
        acc.x += a * v.x; acc.y += a * v.y; acc.z += a * v.z; acc.w += a * v.w;
      }
    }
    acc.x += bias[c4 + 0]; acc.y += bias[c4 + 1];
    acc.z += bias[c4 + 2]; acc.w += bias[c4 + 3];
    if (RELU) {
      acc.x = fmaxf(acc.x, 0.f); acc.y = fmaxf(acc.y, 0.f);
      acc.z = fmaxf(acc.z, 0.f); acc.w = fmaxf(acc.w, 0.f);
    }
    *reinterpret_cast<float4*>(&s_acc[n * CH + c4]) = acc;
  }
  __syncthreads();

  // 4) GraphNorm: xc = x - ms*mean; out = w*xc*rsqrt(var+eps) + b
  if (tid < CH) {
    float mean = 0.0f;
    for (int n = 0; n < NPG; ++n) mean += s_acc[n * CH + tid];
    mean *= (1.0f / NPG);
    const float msmean = gms[tid] * mean;
    float var = 0.0f;
    for (int n = 0; n < NPG; ++n) {
      const float xc = s_acc[n * CH + tid] - msmean;
      var += xc * xc;
    }
    var *= (1.0f / NPG);
    s_msmean[tid] = msmean;
    s_rstd[tid]   = rsqrtf(var + GN_EPS);
  }
  __syncthreads();

  if (!FINAL) {
    float4* gout = (float4*)(h_out + (size_t)nodeBase * CH);
    for (int i = tid; i < NPG * CH / 4; i += 256) {
      const int c4 = (i & 15) << 2;
      const float4 v = *reinterpret_cast<const float4*>(&s_acc[i * 4]);
      float4 o;
      o.x = gw[c4 + 0] * (v.x - s_msmean[c4 + 0]) * s_rstd[c4 + 0] + gb[c4 + 0];
      o.y = gw[c4 + 1] * (v.y - s_msmean[c4 + 1]) * s_rstd[c4 + 1] + gb[c4 + 1];
      o.z = gw[c4 + 2] * (v.z - s_msmean[c4 + 2]) * s_rstd[c4 + 2] + gb[c4 + 2];
      o.w = gw[c4 + 3] * (v.w - s_msmean[c4 + 3]) * s_rstd[c4 + 3] + gb[c4 + 3];
      gout[i] = o;
    }
  } else {
    for (int i = tid; i < NPG * CH; i += 256) {
      const int c = i & (CH - 1);
      s_acc[i] = gw[c] * (s_acc[i] - s_msmean[c]) * s_rstd[c] + gb[c];
    }
    __syncthreads();
    if (tid < CH) {
      float p = 0.0f;
      for (int n = 0; n < NPG; ++n) p += s_acc[n * CH + tid];
      p *= (1.0f / NPG);
      s_pool[tid] = p;
      out_pooled[(size_t)g * CH + tid] = p;
    }
    __syncthreads();
    if (tid < 2) {
      float o = linb[tid];
      for (int c = 0; c < CH; ++c) o += s_pool[c] * linW[c * 2 + tid];
      out_o[g * 2 + tid] = o;
    }
  }
}

// ---------------------------------------------------------------------------
extern "C" void kernel_launch(void* const* d_in, const int* in_sizes, int n_in,
                              void* d_out, int out_size, void* d_ws, size_t ws_size,
                              hipStream_t stream)
{
  (void)in_sizes; (void)n_in; (void)out_size; (void)ws_size;

  const float* x     = (const float*)d_in[0];
  const int*   ei    = (const int*)  d_in[1];
  // d_in[2] = batch: unused (graphs are contiguous blocks of NPG nodes)
  const float* W1l   = (const float*)d_in[3];
  const float* W1r   = (const float*)d_in[4];
  const float* att1  = (const float*)d_in[5];
  const float* b1    = (const float*)d_in[6];
  const float* W2l   = (const float*)d_in[7];
  const float* W2r   = (const float*)d_in[8];
  const float* att2  = (const float*)d_in[9];
  const float* b2    = (const float*)d_in[10];
  const float* W3l   = (const float*)d_in[11];
  const float* W3r   = (const float*)d_in[12];
  const float* att3  = (const float*)d_in[13];
  const float* b3    = (const float*)d_in[14];
  const float* gn1w  = (const float*)d_in[15];
  const float* gn1b  = (const float*)d_in[16];
  const float* gn1ms = (const float*)d_in[17];
  const float* gn2w  = (const float*)d_in[18];
  const float* gn2b  = (const float*)d_in[19];
  const float* gn2ms = (const float*)d_in[20];
  const float* gn3w  = (const float*)d_in[21];
  const float* gn3b  = (const float*)d_in[22];
  const float* gn3ms = (const float*)d_in[23];
  const float* linW  = (const float*)d_in[24];
  const float* linb  = (const float*)d_in[25];

  const int* e_src = ei;
  const int* e_dst = ei + E_;

  float* ws = (float*)d_ws;
  float* xl = ws;
  float* xr = ws + (size_t)NTOT * CH;
  float* h  = ws + (size_t)2 * NTOT * CH;

  float* out_o      = (float*)d_out;        // [G,2] first in return order
  float* out_pooled = out_o + G_ * 2;       // then [G,C]

  const dim3 gGrid(NTOT / 32), gBlk(32);    // 32-row tiles per wave
  const dim3 aGrid(G_),        aBlk(256);

  // Layer 1 (K = FIN = 22, zero-padded in registers)
  wmma_gemm_n64<FIN><<<gGrid, gBlk, 0, stream>>>(x, W1l, xl);
  wmma_gemm_n64<FIN><<<gGrid, gBlk, 0, stream>>>(x, W1r, xr);
  gat_graph_kernel<1, 0><<<aGrid, aBlk, 0, stream>>>(
      xl, xr, e_src, e_dst, att1, b1, gn1w, gn1b, gn1ms,
      h, nullptr, nullptr, nullptr, nullptr);

  // Layer 2
  wmma_gemm_n64<CH><<<gGrid, gBlk, 0, stream>>>(h, W2l, xl);
  wmma_gemm_n64<CH><<<gGrid, gBlk, 0, stream>>>(h, W2r, xr);
  gat_graph_kernel<1, 0><<<aGrid, aBlk, 0, stream>>>(
      xl, xr, e_src, e_dst, att2, b2, gn2w, gn2b, gn2ms,
      h, nullptr, nullptr, nullptr, nullptr);

  // Layer 3 (no ReLU) + GraphNorm + mean-pool + classifier
  wmma_gemm_n64<CH><<<gGrid, gBlk, 0, stream>>>(h, W3l, xl);
  wmma_gemm_n64<CH><<<gGrid, gBlk, 0, stream>>>(h, W3r, xr);
  gat_graph_kernel<0, 1><<<aGrid, aBlk, 0, stream>>>(
      xl, xr, e_src, e_dst, att3, b3, gn3w, gn3b, gn3ms,
      nullptr, linW, linb, out_o, out_pooled);
}